// AvgPool2d_8993661518235
// MI455X (gfx1250) — compile-verified
//
#include <hip/hip_runtime.h>

// C[64,N] = A[64,K] * W[N,K]^T using V_WMMA_F32_16X16X4_F32 (exact f32 math).
// HBM-streaming bound (pad_mat 256MB / weight 64MB read once; A is L2-resident).
// Staging uses CDNA5 GLOBAL_LOAD_ASYNC_TO_LDS_B128 + ASYNCcnt with
// double-buffered LDS tiles so tile t+1 streams while tile t runs WMMA.

typedef __attribute__((ext_vector_type(2))) float v2f;
typedef __attribute__((ext_vector_type(8))) float v8f;

#define TK 64          // K tile staged through LDS
#define LDSW 68        // 64 + 4 pad: conflict-free ds_load_b64 fragment reads
#define BM 64          // full batch (M == 64 for both stages)
#define BN 32          // N tile per workgroup (2 wmma tiles wide)
#define NTHREADS 256   // 8 waves: 4 M-tiles x 2 N-tiles

__device__ __forceinline__ void async_copy_b128(unsigned lds_off, const void* g) {
    // VDST = per-lane LDS byte offset, VADDR = per-lane 64-bit global address.
    asm volatile("global_load_async_to_lds_b128 %0, %1, off"
                 :: "v"(lds_off), "v"(g) : "memory");
}
__device__ __forceinline__ void wait_async0() {
    asm volatile("s_wait_asynccnt 0x0" ::: "memory");
}

__global__ __launch_bounds__(NTHREADS)
void gemm_abT_f32_wmma(const float* __restrict__ A,  // [64, K] row-major
                       const float* __restrict__ W,  // [N,  K] row-major
                       float* __restrict__ C,        // [64, N] row-major
                       int N, int K) {
    __shared__ float a_lds[2][BM * LDSW];   // 2 x (64 x 64 +pad)
    __shared__ float w_lds[2][BN * LDSW];   // 2 x (32 x 64 +pad)

    const int tid  = threadIdx.x;
    const int lane = tid & 31;
    const int wave = tid >> 5;
    const int n0   = blockIdx.x * BN;

    const int mt    = (wave & 3) * 16;  // wave's M-tile base (global, M==BM)
    const int nt    = (wave >> 2) * 16; // wave's N-tile base within block
    const int lrow  = lane & 15;
    const int khalf = (lane >> 4) * 2;  // lanes 16-31 hold K=2,3 of each step

    // Issue async HBM->LDS copies for one K-tile into buffer `buf`.
    // A tile: 64x64 f32 = 1024 x b128 chunks; W tile: 32x64 = 512 chunks.
    auto stage = [&](int buf, int k0) {
#pragma unroll
        for (int i = 0; i < 4; ++i) {
            int idx = tid + i * NTHREADS;
            int row = idx >> 4;                 // 0..63
            int col = (idx & 15) << 2;          // 0..60 step 4
            async_copy_b128((unsigned)(uintptr_t)&a_lds[buf][row * LDSW + col],
                            A + (size_t)row * K + k0 + col);
        }
#pragma unroll
        for (int i = 0; i < 2; ++i) {
            int idx = tid + i * NTHREADS;
            int row = idx >> 4;                 // 0..31
            int col = (idx & 15) << 2;
            async_copy_b128((unsigned)(uintptr_t)&w_lds[buf][row * LDSW + col],
                            W + (size_t)(n0 + row) * K + k0 + col);
        }
    };

    v8f acc = {0.f, 0.f, 0.f, 0.f, 0.f, 0.f, 0.f, 0.f};

    // Prologue: stream tile 0 into buffer 0.
    stage(0, 0);
    wait_async0();
    __syncthreads();

    const int T = K / TK;
    for (int t = 0; t < T; ++t) {
        const int buf = t & 1;
        if (t + 1 < T) stage(buf ^ 1, (t + 1) * TK);   // overlap next tile

        // 16 x V_WMMA_F32_16X16X4_F32 over the current K tile.
        // A frag (16x4): lane L -> A[M = L&15][K = khalf+{0,1}]
        // B frag (4x16): lane L -> W[N = L&15][K = khalf+{0,1}]
        const float* ap = &a_lds[buf][(mt + lrow) * LDSW + khalf];
        const float* wp = &w_lds[buf][(nt + lrow) * LDSW + khalf];
#pragma unroll
        for (int kk = 0; kk < TK; kk += 4) {
            v2f af = *(const v2f*)(ap + kk);
            v2f bf = *(const v2f*)(wp + kk);
            acc = __builtin_amdgcn_wmma_f32_16x16x4_f32(
                false, af, false, bf, (short)0, acc, false, false);
        }

        wait_async0();       // our wave's async writes for tile t+1 landed
        __syncthreads();     // everyone's writes landed / readers done
    }

    // Store C 16x16 tile: VGPR r -> M = r + 8*(lane>=16), N = lane&15.
    const int n     = n0 + nt + lrow;
    const int mbase = mt + (lane >> 4) * 8;
#pragma unroll
    for (int r = 0; r < 8; ++r) {
        C[(size_t)(mbase + r) * N + n] = acc[r];
    }
}

extern "C" void kernel_launch(void* const* d_in, const int* in_sizes, int n_in,
                              void* d_out, int out_size, void* d_ws, size_t ws_size,
                              hipStream_t stream) {
    const float* enc_x   = (const float*)d_in[0];  // [64, D]
    const float* weight  = (const float*)d_in[1];  // [Nout, D]
    const float* pad_mat = (const float*)d_in[2];  // [D, D]
    // d_in[3] (inv_pad_mat) unused by the reference forward path.

    const int B    = 64;
    const int D    = in_sizes[0] / B;      // 8192
    const int Nout = out_size / B;         // 2048

    float* padded = (float*)d_ws;          // [64, D] intermediate (2 MB)
    float* out    = (float*)d_out;         // [64, Nout]

    // Stage 1: padded = enc_x @ pad_mat^T
    gemm_abT_f32_wmma<<<dim3(D / BN), dim3(NTHREADS), 0, stream>>>(
        enc_x, pad_mat, padded, D, D);

    // Stage 2: out = padded @ weight^T
    gemm_abT_f32_wmma<<<dim3(Nout / BN), dim3(NTHREADS), 0, stream>>>(
        padded, weight, out, Nout, D);
}